// Transformer_103079215399
// MI455X (gfx1250) — compile-verified
//
#include <hip/hip_runtime.h>
#include <math.h>

typedef __attribute__((ext_vector_type(2))) float v2f;
typedef __attribute__((ext_vector_type(8))) float v8f;

#define DDIM 512
#define BR 32          // Q rows per workgroup
#define BS 64          // KV rows per stream block
#define QST 516        // padded LDS stride for Q/KV rows (516 % 64 == 4 -> conflict-free)
#define SST 66         // padded LDS stride for score tile rows
#define NTHREADS 256   // 8 waves

// ---------------------------------------------------------------------------
// CDNA5 async global->LDS copy (ASYNCcnt path, no VGPR round trip).
// Generic pointers to LDS carry the LDS byte offset in addr[31:0] (ISA 10.2),
// which is exactly the VDST operand of GLOBAL_LOAD_ASYNC_TO_LDS.
// ---------------------------------------------------------------------------
__device__ __forceinline__ void async_b128_to_lds(void* lds_ptr, const void* gptr)
{
  unsigned lds_off = (unsigned)(uintptr_t)lds_ptr;            // LDS byte offset
  unsigned long long gaddr = (unsigned long long)(uintptr_t)gptr;
  asm volatile("global_load_async_to_lds_b128 %0, %1, off"
               :
               : "v"(lds_off), "v"(gaddr)
               : "memory");
}

__device__ __forceinline__ void wait_async0()
{
  asm volatile("s_wait_asynccnt 0" ::: "memory");
}

// ---------------------------------------------------------------------------
// Kernel 0: c_max = max over all rows (of both matrices) of squared L2 norm.
// One wave per row; positive floats compare correctly as ints -> atomicMax.
// ---------------------------------------------------------------------------
__launch_bounds__(NTHREADS)
__global__ void cmax_kernel(const float* __restrict__ XB,
                            const float* __restrict__ XBar,
                            int N, int M, int* __restrict__ out_bits)
{
  const int gtid = blockIdx.x * NTHREADS + threadIdx.x;
  const int wrow = gtid >> 5;            // global wave id == row id
  const int lane = threadIdx.x & 31;
  if (wrow >= N + M) return;
  const float* row = (wrow < N) ? (XB + (size_t)wrow * DDIM)
                                : (XBar + (size_t)(wrow - N) * DDIM);
  float ss = 0.0f;
#pragma unroll
  for (int i = 0; i < DDIM / 32; ++i) {
    float x = row[lane + i * 32];
    ss = fmaf(x, x, ss);
  }
#pragma unroll
  for (int off = 16; off; off >>= 1) ss += __shfl_xor(ss, off, 32);
  if (lane == 0) atomicMax(out_bits, __float_as_int(ss));
}

// ---------------------------------------------------------------------------
// Fused attention pass (used twice with Q/KV roles swapped):
//   O[r,:] = sum_s exp(scale * <Q_r, KV_s>) * KV[s,:]  /  sum_s exp(...)
// scale = 1 / (sqrt(D) * c_max).  C is never materialized in HBM.
// ---------------------------------------------------------------------------
__launch_bounds__(NTHREADS)
__global__ void attn_kernel(const float* __restrict__ Q,
                            const float* __restrict__ KV,
                            float* __restrict__ O,
                            int S, const int* __restrict__ cmax_bits)
{
  extern __shared__ float smem[];
  float* q_lds  = smem;                                   // [BR][QST]
  float* kv_lds = smem + BR * QST;                        // [BS][QST]
  float* s_lds  = smem + BR * QST + BS * QST;             // [BR][SST]
  float* rowsum = smem + BR * QST + BS * QST + BR * SST;  // [BR]

  const int tid  = threadIdx.x;
  const int w    = tid >> 5;       // wave 0..7
  const int lane = tid & 31;
  const int ln   = lane & 15;      // N / M index within 16
  const int lh   = lane >> 4;      // lane half (K split for A/B frags)
  const int row0 = blockIdx.x * BR;

  const float cmax  = __int_as_float(*cmax_bits);
  const float scale = 1.0f / (sqrtf((float)DDIM) * cmax);

  // ---- stage Q block into LDS via async copies (16B granules) ----
  {
    const float* Qg = Q + (size_t)row0 * DDIM;
    for (int idx = tid; idx < BR * (DDIM / 4); idx += NTHREADS) {
      int r = idx >> 7;                 // DDIM/4 == 128 float4 per row
      int c = (idx & 127) << 2;
      async_b128_to_lds(&q_lds[r * QST + c], &Qg[(size_t)r * DDIM + c]);
    }
  }
  if (tid < BR) rowsum[tid] = 0.0f;

  // Output accumulators: wave owns row-tile rt_o = w&1, col-tiles ct = 4*j + (w>>1)
  const int rt_o = w & 1;
  const int cg   = w >> 1;
  v8f o_acc[8];
#pragma unroll
  for (int j = 0; j < 8; ++j) o_acc[j] = (v8f)0.0f;

  // Score tile assignment: 32x64 scores = 2x4 tiles of 16x16, one per wave
  const int rt_s = w >> 2;
  const int ct_s = w & 3;

  for (int sb = 0; sb < S; sb += BS) {
    __syncthreads();                    // prior iteration finished reading kv_lds
    // ---- stage KV block [BS][DDIM] into LDS via async copies ----
    {
      const float* Kg = KV + (size_t)sb * DDIM;
      for (int idx = tid; idx < BS * (DDIM / 4); idx += NTHREADS) {
        int r = idx >> 7;
        int c = (idx & 127) << 2;
        async_b128_to_lds(&kv_lds[r * QST + c], &Kg[(size_t)r * DDIM + c]);
      }
    }
    wait_async0();                      // this wave's async copies landed in LDS
    __syncthreads();                    // everyone's copies landed

    // ---- scores: S_tile(rt_s, ct_s) = Q_rows x KV_rows^T over K = DDIM ----
    v8f sacc = (v8f)0.0f;
    const float* qrow = &q_lds[(rt_s * 16 + ln) * QST + 2 * lh];
    const float* krow = &kv_lds[(ct_s * 16 + ln) * QST + 2 * lh];
#pragma unroll 8
    for (int k0 = 0; k0 < DDIM; k0 += 4) {
      v2f a = *(const v2f*)(qrow + k0);   // A[m][k]: K = k0 + 2*lh + {0,1}
      v2f b = *(const v2f*)(krow + k0);   // B[k][n] = KV[n][k] (scores vs KV^T)
      sacc = __builtin_amdgcn_wmma_f32_16x16x4_f32(false, a, false, b,
                                                   (short)0, sacc, false, false);
    }
    // exp() and spill the 16x16 exp-score tile to LDS (C layout: M = v + 8*lh)
#pragma unroll
    for (int v = 0; v < 8; ++v) {
      float e = __expf(sacc[v] * scale);
      s_lds[(rt_s * 16 + 8 * lh + v) * SST + ct_s * 16 + ln] = e;
    }
    __syncthreads();

    // ---- row sums of exp-scores (single writer per row) ----
    if (tid < BR) {
      float rs = 0.0f;
#pragma unroll 8
      for (int j2 = 0; j2 < BS; ++j2) rs += s_lds[tid * SST + j2];
      rowsum[tid] += rs;
    }

    // ---- O += S_exp @ KV_block : 8 owned 16x16 output tiles per wave ----
#pragma unroll
    for (int j = 0; j < 8; ++j) {
      const int ct = 4 * j + cg;          // output col-tile (d dimension)
      v8f acc = o_acc[j];
#pragma unroll
      for (int kk = 0; kk < BS; kk += 4) {
        v2f a = *(const v2f*)&s_lds[(rt_o * 16 + ln) * SST + kk + 2 * lh];
        v2f b;
        b.x = kv_lds[(kk + 2 * lh + 0) * QST + ct * 16 + ln];
        b.y = kv_lds[(kk + 2 * lh + 1) * QST + ct * 16 + ln];
        acc = __builtin_amdgcn_wmma_f32_16x16x4_f32(false, a, false, b,
                                                    (short)0, acc, false, false);
      }
      o_acc[j] = acc;
    }
  }
  __syncthreads();

  // ---- normalize by row sums and store ----
  float rinv[8];
#pragma unroll
  for (int v = 0; v < 8; ++v) rinv[v] = 1.0f / rowsum[rt_o * 16 + 8 * lh + v];
#pragma unroll
  for (int j = 0; j < 8; ++j) {
    const int ct = 4 * j + cg;
#pragma unroll
    for (int v = 0; v < 8; ++v) {
      int r = rt_o * 16 + 8 * lh + v;
      O[(size_t)(row0 + r) * DDIM + ct * 16 + ln] = o_acc[j][v] * rinv[v];
    }
  }
}

// ---------------------------------------------------------------------------
extern "C" void kernel_launch(void* const* d_in, const int* in_sizes, int n_in,
                              void* d_out, int out_size, void* d_ws, size_t ws_size,
                              hipStream_t stream)
{
  const float* XB   = (const float*)d_in[0];
  const float* XBar = (const float*)d_in[1];
  const int N = in_sizes[0] / DDIM;   // 32768
  const int M = in_sizes[1] / DDIM;   // 4096
  float* out = (float*)d_out;

  // ws[0] holds c_max as float bits; reset every call (capturable memset).
  hipMemsetAsync(d_ws, 0, sizeof(int), stream);

  const int rows = N + M;
  const int cmax_blocks = (rows * 32 + NTHREADS - 1) / NTHREADS;
  cmax_kernel<<<cmax_blocks, NTHREADS, 0, stream>>>(XB, XBar, N, M, (int*)d_ws);

  const size_t lds_bytes = (size_t)(BR * QST + BS * QST + BR * SST + BR) * sizeof(float);

  // Pass 1: rows of X_B attend over X_bar -> d_out[0 .. N*D)
  attn_kernel<<<N / BR, NTHREADS, lds_bytes, stream>>>(
      XB, XBar, out, M, (const int*)d_ws);

  // Pass 2: rows of X_bar attend over X_B -> d_out[N*D .. (N+M)*D)
  attn_kernel<<<M / BR, NTHREADS, lds_bytes, stream>>>(
      XBar, XB, out + (size_t)N * DDIM, N, (const int*)d_ws);
}